// MeanAggregator_89275190215130
// MI455X (gfx1250) — compile-verified
//
#include <hip/hip_runtime.h>
#include <stdint.h>

// Masked mean of neighbor embeddings (gather-reduce), MI455X / gfx1250.
//
// features : [N_NODES=100000, D=128] f32   (51.2 MB -> resident in 192MB L2)
// neigh_idx: [B=50000, K=25] int64 (values < 2^31 -> low dword suffices)
// neigh_mask:[B=50000, K=25] bool (1 byte)
// out      : [B, D] f32
//
// One wave32 per output row.
//  1) Lanes 0..24 fetch the row's 25 indices (low dwords) and mask bytes with
//     two coalesced vector loads.
//  2) ballot_w32 -> 25 mask bits in one SGPR; popcount -> divisor count.
//  3) Walk set bits (scalar ctz loop, EXEC stays full); v_readlane moves each
//     live index to an SGPR; issue compacted CDNA5 async row gathers
//     (global_load_async_to_lds_b128: 32 lanes x 16B = one 512B feature row,
//     up to 25 outstanding on ASYNCcnt).
//  4) s_wait_asynccnt 0, drain rows from LDS (ds_load_b128), IEEE-divide by
//     max(cnt,1), non-temporal store (keep features resident in L2).

typedef float v4f __attribute__((ext_vector_type(4)));

#define D_FEAT 128
#define K_NEIGH 25
#define WAVES_PER_BLK 4

__global__ __launch_bounds__(WAVES_PER_BLK * 32)
void mean_agg_async_kernel(const float* __restrict__ feat,
                           const long long* __restrict__ nidx,
                           const unsigned char* __restrict__ nmask,
                           float* __restrict__ out,
                           int batch)
{
    // 4 waves * 25 rows * 512B = 51200 B static LDS
    __shared__ __align__(16) float sh[WAVES_PER_BLK][K_NEIGH][D_FEAT];

    const int lane = threadIdx.x & 31;
    const int wv   = threadIdx.x >> 5;

    int b = blockIdx.x * WAVES_PER_BLK + wv;
    b = __builtin_amdgcn_readfirstlane(b);   // wave-uniform -> scalar regs
    if (b >= batch) return;

    const long long*     ip = nidx  + (long long)b * K_NEIGH;
    const unsigned char* mp = nmask + (long long)b * K_NEIGH;

    // Lane-parallel fetch of this row's metadata (2 coalesced vector loads
    // instead of 25 serialized s_load_u8 round trips).
    int idx_lo = 0;
    int mym    = 0;
    if (lane < K_NEIGH) {
        idx_lo = ((const int*)ip)[lane * 2];   // little-endian low dword
        mym    = mp[lane];
    }

    // 25 mask bits -> one SGPR.
    const unsigned live = __builtin_amdgcn_ballot_w32(mym != 0) &
                          ((1u << K_NEIGH) - 1u);
    const int cnt = __builtin_popcount(live);

    // Per-lane LDS byte address: low 32 bits of the generic pointer are the
    // workgroup-relative LDS offset (FLAT aperture layout, ISA ch.10.2).
    float* lbase = &sh[wv][0][lane * 4];
    const unsigned laddr0 = (unsigned)(unsigned long long)lbase;

    // Issue compacted async row gathers for the live neighbors.
    unsigned rem  = live;
    int      slot = 0;
    while (rem) {
        const int k = __builtin_ctz(rem);
        rem &= rem - 1u;
        const int idxk = __builtin_amdgcn_readlane(idx_lo, k);  // SGPR index
        const float* gp = feat + (long long)idxk * D_FEAT + lane * 4;
        const unsigned la = laddr0 + (unsigned)(slot * (D_FEAT * 4));
        asm volatile("global_load_async_to_lds_b128 %0, %1, off"
                     :: "v"(la), "v"(gp)
                     : "memory");
        ++slot;
    }

    // Wait for all async row fetches to land in LDS (ASYNCcnt -> 0).
    __builtin_amdgcn_s_wait_asynccnt(0);
    asm volatile("" ::: "memory");  // keep LDS reads below the wait

    // Drain LDS: each lane owns 4 consecutive floats of the 128-wide row.
    v4f acc = {0.0f, 0.0f, 0.0f, 0.0f};
    const v4f* sp = (const v4f*)lbase;
    for (int r = 0; r < cnt; ++r) {
        acc += sp[r * (D_FEAT / 4)];   // ds_load_b128, conflict-free
    }

    // Reference: summed / max(count, 1)  (exact IEEE divide)
    const float c = (float)(cnt > 0 ? cnt : 1);
    v4f res;
    res.x = acc.x / c;
    res.y = acc.y / c;
    res.z = acc.z / c;
    res.w = acc.w / c;

    // Output written once -> non-temporal so features stay resident in L2.
    v4f* op = (v4f*)(out + (long long)b * D_FEAT + lane * 4);
    __builtin_nontemporal_store(res, op);
}

extern "C" void kernel_launch(void* const* d_in, const int* in_sizes, int n_in,
                              void* d_out, int out_size, void* d_ws, size_t ws_size,
                              hipStream_t stream) {
    const float*         feat  = (const float*)d_in[0];
    const long long*     nidx  = (const long long*)d_in[1];
    const unsigned char* nmask = (const unsigned char*)d_in[2];
    float*               out   = (float*)d_out;

    const int batch = in_sizes[1] / K_NEIGH;   // 50000
    const int blocks = (batch + WAVES_PER_BLK - 1) / WAVES_PER_BLK;

    mean_agg_async_kernel<<<blocks, WAVES_PER_BLK * 32, 0, stream>>>(
        feat, nidx, nmask, out, batch);
}